// HyperKGL_38646115729707
// MI455X (gfx1250) — compile-verified
//
#include <hip/hip_runtime.h>
#include <math.h>

typedef __attribute__((ext_vector_type(2))) float v2f;
typedef __attribute__((ext_vector_type(8))) float v8f;

#define EPSF 1e-9f

// ---------- monotone float<->uint key for atomic segment-max ----------
__device__ __forceinline__ unsigned fkey(float f) {
    unsigned u = __float_as_uint(f);
    return (u & 0x80000000u) ? ~u : (u | 0x80000000u);
}
__device__ __forceinline__ float finv(unsigned k) {
    unsigned u = (k & 0x80000000u) ? (k & 0x7FFFFFFFu) : ~k;
    return __uint_as_float(u);
}

// ---------- generic WMMA f32 GEMM: Y = act( X1@W1 [+ X2@W2 [+ X3@W3]] ) ----------
// X*: (R x 64) row-major f32, W*: (64 x 64) row-major f32, Y: (R x 64).
// act: 0=none, 1=leaky_relu(0.01), 2=tanh
// Block = 256 threads = 8 waves; one 16-row output tile per wave.
// W is staged into LDS pre-swizzled into B-fragment order so each lane's
// fragment is one aligned ds_load_b64:
//   W[kr][n] -> wl[j][ (kt*2 + hi)*128 + n*2 + pos ],
//   kt = kr>>2 (k-step), hi = (kr&3)>>1 (half-wave), pos = kr&1 (.x/.y)
__global__ void gemm3_kernel(const float* __restrict__ X1, const float* __restrict__ W1,
                             const float* __restrict__ X2, const float* __restrict__ W2,
                             const float* __restrict__ X3, const float* __restrict__ W3,
                             float* __restrict__ Y, int R, int act)
{
    __shared__ float wl[3][64 * 64];
    const float* Ws[3] = {W1, W2, W3};
    const float* Xs[3] = {X1, X2, X3};
    for (int j = 0; j < 3; ++j) {
        const float* W = Ws[j];
        if (!W) continue;
        for (int idx = threadIdx.x; idx < 64 * 64; idx += blockDim.x) {
            const int kr = idx >> 6;
            const int n  = idx & 63;
            const int dst = ((kr >> 2) * 2 + ((kr & 3) >> 1)) * 128 + n * 2 + (kr & 1);
            wl[j][dst] = W[idx];
        }
    }
    __syncthreads();

    const int wave = threadIdx.x >> 5;
    const int lane = threadIdx.x & 31;
    const int row0 = (blockIdx.x * 8 + wave) * 16;
    if (row0 >= R) return;

    const int l15 = lane & 15;
    const int hi  = lane >> 4;   // 0 or 1

    const int   row   = row0 + l15;
    const float amask = (row < R) ? 1.0f : 0.0f;
    const int   rowc  = (row < R) ? row : (R - 1);

    v8f acc0 = {}, acc1 = {}, acc2 = {}, acc3 = {};

    for (int j = 0; j < 3; ++j) {
        const float* X = Xs[j];
        if (!X) continue;
        const float* xrow = X + (size_t)rowc * 64 + hi * 2;
        const float* wj   = &wl[j][hi * 128 + l15 * 2];
        #pragma unroll
        for (int kt = 0; kt < 16; ++kt) {
            v2f a = *(const v2f*)(xrow + kt * 4);   // unconditional global_load_b64
            a.x *= amask;
            a.y *= amask;
            const float* bp = wj + kt * 256;
            v2f b0 = *(const v2f*)(bp);             // ds_load_b64, no repack
            v2f b1 = *(const v2f*)(bp + 32);
            v2f b2 = *(const v2f*)(bp + 64);
            v2f b3 = *(const v2f*)(bp + 96);
            acc0 = __builtin_amdgcn_wmma_f32_16x16x4_f32(false, a, false, b0, (short)0, acc0, false, false);
            acc1 = __builtin_amdgcn_wmma_f32_16x16x4_f32(false, a, false, b1, (short)0, acc1, false, false);
            acc2 = __builtin_amdgcn_wmma_f32_16x16x4_f32(false, a, false, b2, (short)0, acc2, false, false);
            acc3 = __builtin_amdgcn_wmma_f32_16x16x4_f32(false, a, false, b3, (short)0, acc3, false, false);
        }
    }

    v8f accs[4] = {acc0, acc1, acc2, acc3};
    const int mOff = hi * 8;
    for (int c = 0; c < 4; ++c) {
        #pragma unroll
        for (int v = 0; v < 8; ++v) {
            const int r = row0 + v + mOff;
            if (r < R) {
                float y = accs[c][v];
                if (act == 1) y = (y > 0.0f) ? y : 0.01f * y;
                else if (act == 2) y = tanhf(y);
                Y[(size_t)r * 64 + c * 16 + l15] = y;
            }
        }
    }
}

// ---------- utility fills ----------
__global__ void fill_f32_kernel(float* __restrict__ p, float v, long long n) {
    long long i = (long long)blockIdx.x * blockDim.x + threadIdx.x;
    const long long stride = (long long)gridDim.x * blockDim.x;
    for (; i < n; i += stride) p[i] = v;
}
__global__ void fill_u32_kernel(unsigned* __restrict__ p, unsigned v, long long n) {
    long long i = (long long)blockIdx.x * blockDim.x + threadIdx.x;
    const long long stride = (long long)gridDim.x * blockDim.x;
    for (; i < n; i += stride) p[i] = v;
}
__global__ void set_member_kernel(int* __restrict__ member, const int* __restrict__ idx, int nb) {
    int i = blockIdx.x * blockDim.x + threadIdx.x;
    if (i < nb) member[idx[i]] = 1;
}

// ---------- edge pass 1: logits + segment max ----------
// 16 lanes per edge, float4 per lane.
__global__ void edge_logits_kernel(const int* __restrict__ sl, const int* __restrict__ member,
                                   const float* __restrict__ mn, const float* __restrict__ ms,
                                   const float* __restrict__ qh,
                                   float* __restrict__ logit, unsigned* __restrict__ lmaxkey, int E)
{
    long long gid = (long long)blockIdx.x * blockDim.x + threadIdx.x;
    int e = (int)(gid >> 4);
    int l = (int)(gid & 15);
    if (e >= E) return;
    __builtin_prefetch(sl + 3 * e + 2048, 0, 0);   // stream the link table ahead
    int src = sl[3 * e], hed = sl[3 * e + 1], sem = sl[3 * e + 2];
    float4 a = ((const float4*)(mn + (size_t)src * 64))[l];
    float4 s = ((const float4*)(ms + (size_t)sem * 64))[l];
    float4 q = ((const float4*)(qh + (size_t)hed * 64))[l];
    float part = (a.x + s.x) * q.x + (a.y + s.y) * q.y + (a.z + s.z) * q.z + (a.w + s.w) * q.w;
    part += __shfl_xor(part, 8, 32);
    part += __shfl_xor(part, 4, 32);
    part += __shfl_xor(part, 2, 32);
    part += __shfl_xor(part, 1, 32);
    float lg = member[src] ? part : -1e30f;
    if (l == 0) {
        logit[e] = lg;
        atomicMax(lmaxkey + hed, fkey(lg));
    }
}

// ---------- edge pass 2: softmax weights, weighted segment sums ----------
__global__ void edge_accum_kernel(const int* __restrict__ sl, const int* __restrict__ member,
                                  const float* __restrict__ mn, const float* __restrict__ ms,
                                  const float* __restrict__ logit, const unsigned* __restrict__ lmaxkey,
                                  float* __restrict__ denom, float* __restrict__ cnt,
                                  float* __restrict__ agg, int E)
{
    long long gid = (long long)blockIdx.x * blockDim.x + threadIdx.x;
    int e = (int)(gid >> 4);
    int l = (int)(gid & 15);
    if (e >= E) return;
    int src = sl[3 * e], hed = sl[3 * e + 1], sem = sl[3 * e + 2];
    if (!member[src]) return;
    float w = expf(logit[e] - finv(lmaxkey[hed]));
    if (l == 0) {
        atomicAdd(denom + hed, w);
        atomicAdd(cnt + hed, 1.0f);
    }
    float4 a = ((const float4*)(mn + (size_t)src * 64))[l];
    float4 s = ((const float4*)(ms + (size_t)sem * 64))[l];
    float* ap = agg + (size_t)hed * 64 + l * 4;
    atomicAdd(ap + 0, w * (a.x + s.x));
    atomicAdd(ap + 1, w * (a.y + s.y));
    atomicAdd(ap + 2, w * (a.z + s.z));
    atomicAdd(ap + 3, w * (a.w + s.w));
}

// ---------- normalize agg by denom ----------
__global__ void normalize_h_kernel(float* __restrict__ agg, const float* __restrict__ denom, long long n) {
    long long i = (long long)blockIdx.x * blockDim.x + threadIdx.x;
    if (i < n) agg[i] = agg[i] / (denom[i >> 6] + EPSF);
}

// ---------- he = cnt>0 ? he_tmp : he ----------
__global__ void select_he_kernel(float* __restrict__ he, const float* __restrict__ he_tmp,
                                 const float* __restrict__ cnt, long long n) {
    long long i = (long long)blockIdx.x * blockDim.x + threadIdx.x;
    if (i < n && cnt[i >> 6] > 0.0f) he[i] = he_tmp[i];
}

// ---------- edge pass 3: scatter he[hed] (and he*sema) back to src nodes ----------
__global__ void edge_node_kernel(const int* __restrict__ sl, const int* __restrict__ member,
                                 const float* __restrict__ he, const float* __restrict__ sema,
                                 float* __restrict__ n_sum, float* __restrict__ o_sum,
                                 float* __restrict__ n_cnt, int E)
{
    long long gid = (long long)blockIdx.x * blockDim.x + threadIdx.x;
    int e = (int)(gid >> 4);
    int l = (int)(gid & 15);
    if (e >= E) return;
    int src = sl[3 * e], hed = sl[3 * e + 1], sem = sl[3 * e + 2];
    if (!member[src]) return;
    if (l == 0) atomicAdd(n_cnt + src, 1.0f);
    float4 h = ((const float4*)(he + (size_t)hed * 64))[l];
    float4 s = ((const float4*)(sema + (size_t)sem * 64))[l];
    float* np = n_sum + (size_t)src * 64 + l * 4;
    float* op = o_sum + (size_t)src * 64 + l * 4;
    atomicAdd(np + 0, h.x);        atomicAdd(np + 1, h.y);
    atomicAdd(np + 2, h.z);        atomicAdd(np + 3, h.w);
    atomicAdd(op + 0, h.x * s.x);  atomicAdd(op + 1, h.y * s.y);
    atomicAdd(op + 2, h.z * s.z);  atomicAdd(op + 3, h.w * s.w);
}

// ---------- normalize node sums by counts (in place) ----------
__global__ void normalize_n_kernel(float* __restrict__ n_sum, float* __restrict__ o_sum,
                                   const float* __restrict__ n_cnt, long long n) {
    long long i = (long long)blockIdx.x * blockDim.x + threadIdx.x;
    if (i < n) {
        float inv = 1.0f / (n_cnt[i >> 6] + EPSF);
        n_sum[i] *= inv;
        o_sum[i] *= inv;
    }
}

// ---------- final scatter: out rows at node_indices <- ne rows ----------
__global__ void scatter_out_kernel(float* __restrict__ out, const float* __restrict__ ne,
                                   const int* __restrict__ idx, int nb) {
    long long gid = (long long)blockIdx.x * blockDim.x + threadIdx.x;
    int j = (int)(gid >> 6);
    int d = (int)(gid & 63);
    if (j < nb) {
        int r = idx[j];
        out[(size_t)r * 64 + d] = ne[(size_t)r * 64 + d];
    }
}

static inline unsigned ceil_div_u(long long a, long long b) { return (unsigned)((a + b - 1) / b); }

extern "C" void kernel_launch(void* const* d_in, const int* in_sizes, int n_in,
                              void* d_out, int out_size, void* d_ws, size_t ws_size,
                              hipStream_t stream)
{
    const float* node_emb  = (const float*)d_in[0];
    const float* hyper_emb = (const float*)d_in[1];
    const float* sema_emb  = (const float*)d_in[2];
    const float* W_msg     = (const float*)d_in[3];
    const float* W_att     = (const float*)d_in[4];
    const float* W_upd     = (const float*)d_in[5];
    const float* Wi1       = (const float*)d_in[6];
    const float* Wi2       = (const float*)d_in[7];
    const float* Wo_n      = (const float*)d_in[8];
    const float* Wo_e      = (const float*)d_in[9];
    const float* Wo_s      = (const float*)d_in[10];
    const int* node_indices = (const int*)d_in[11];
    const int* semalinks    = (const int*)d_in[14];

    const int N  = in_sizes[0] / 64;
    const int H  = in_sizes[1] / 64;
    const int S  = in_sizes[2] / 64;
    const int L  = in_sizes[3] / 4096;
    const int NB = in_sizes[11];
    const int E  = in_sizes[14] / 3;

    // ---- carve workspace ----
    char* p = (char*)d_ws;
    auto carve = [&](size_t bytes) -> void* {
        void* r = (void*)p;
        p += (bytes + 255) & ~(size_t)255;
        return r;
    };
    float* neA      = (float*)carve((size_t)N * 64 * 4);
    float* neB      = (float*)carve((size_t)N * 64 * 4);
    float* he       = (float*)carve((size_t)H * 64 * 4);
    float* he_tmp   = (float*)carve((size_t)H * 64 * 4);
    float* mn       = (float*)carve((size_t)N * 64 * 4);
    float* qh       = (float*)carve((size_t)H * 64 * 4);
    float* ms       = (float*)carve((size_t)16 * 64 * 4);
    float* logit    = (float*)carve((size_t)E * 4);
    unsigned* lmaxk = (unsigned*)carve((size_t)H * 4);
    float* denom    = (float*)carve((size_t)H * 4);
    float* cnt_h    = (float*)carve((size_t)H * 4);
    float* agg      = (float*)carve((size_t)H * 64 * 4);
    float* n_sum    = (float*)carve((size_t)N * 64 * 4);
    float* o_sum    = (float*)carve((size_t)N * 64 * 4);
    float* n_cnt    = (float*)carve((size_t)N * 4);
    float* ne_inner = (float*)carve((size_t)N * 64 * 4);
    int* member     = (int*)carve((size_t)N * 4);

    const int TB = 256;
    const unsigned FILLB = 2048;
    auto gemm_grid = [&](int R) { return ceil_div_u(R, 16 * 8); };
    const unsigned edgeGrid = ceil_div_u((long long)E * 16, TB);

    // ---- init: ne/he working copies, membership mask ----
    hipMemcpyAsync(neA, node_emb, (size_t)N * 64 * 4, hipMemcpyDeviceToDevice, stream);
    hipMemcpyAsync(he, hyper_emb, (size_t)H * 64 * 4, hipMemcpyDeviceToDevice, stream);
    fill_u32_kernel<<<FILLB, TB, 0, stream>>>((unsigned*)member, 0u, (long long)N);
    set_member_kernel<<<ceil_div_u(NB, TB), TB, 0, stream>>>(member, node_indices, NB);

    float* ne = neA;
    float* ne_next = neB;

    for (int i = 0; i < L; ++i) {
        const float* Wm   = W_msg + (size_t)i * 4096;
        const float* Wa   = W_att + (size_t)i * 4096;
        const float* Wu   = W_upd + (size_t)i * 4096;
        const float* Wi1i = Wi1 + (size_t)i * 4096;
        const float* Wi2i = Wi2 + (size_t)i * 4096;
        const float* Won  = Wo_n + (size_t)i * 4096;
        const float* Woe  = Wo_e + (size_t)i * 4096;
        const float* Wos  = Wo_s + (size_t)i * 4096;

        // mn = ne @ W_msg ; ms = sema @ W_msg ; qh = he @ W_att
        gemm3_kernel<<<gemm_grid(N), TB, 0, stream>>>(ne, Wm, nullptr, nullptr, nullptr, nullptr, mn, N, 0);
        gemm3_kernel<<<gemm_grid(S), TB, 0, stream>>>(sema_emb, Wm, nullptr, nullptr, nullptr, nullptr, ms, S, 0);
        gemm3_kernel<<<gemm_grid(H), TB, 0, stream>>>(he, Wa, nullptr, nullptr, nullptr, nullptr, qh, H, 0);

        // segment-softmax accumulators
        fill_u32_kernel<<<FILLB, TB, 0, stream>>>(lmaxk, 0x007FFFFFu /* fkey(-inf) */, (long long)H);
        fill_f32_kernel<<<FILLB, TB, 0, stream>>>(denom, 0.0f, (long long)H);
        fill_f32_kernel<<<FILLB, TB, 0, stream>>>(cnt_h, 0.0f, (long long)H);
        fill_f32_kernel<<<FILLB, TB, 0, stream>>>(agg, 0.0f, (long long)H * 64);

        edge_logits_kernel<<<edgeGrid, TB, 0, stream>>>(semalinks, member, mn, ms, qh, logit, lmaxk, E);
        edge_accum_kernel<<<edgeGrid, TB, 0, stream>>>(semalinks, member, mn, ms, logit, lmaxk,
                                                       denom, cnt_h, agg, E);
        normalize_h_kernel<<<ceil_div_u((long long)H * 64, TB), TB, 0, stream>>>(agg, denom, (long long)H * 64);

        // he_tmp = tanh(agg @ W_upd); he = cnt>0 ? he_tmp : he
        gemm3_kernel<<<gemm_grid(H), TB, 0, stream>>>(agg, Wu, nullptr, nullptr, nullptr, nullptr, he_tmp, H, 2);
        select_he_kernel<<<ceil_div_u((long long)H * 64, TB), TB, 0, stream>>>(he, he_tmp, cnt_h, (long long)H * 64);

        // node-side aggregation
        fill_f32_kernel<<<FILLB, TB, 0, stream>>>(n_sum, 0.0f, (long long)N * 64);
        fill_f32_kernel<<<FILLB, TB, 0, stream>>>(o_sum, 0.0f, (long long)N * 64);
        fill_f32_kernel<<<FILLB, TB, 0, stream>>>(n_cnt, 0.0f, (long long)N);
        edge_node_kernel<<<edgeGrid, TB, 0, stream>>>(semalinks, member, he, sema_emb,
                                                      n_sum, o_sum, n_cnt, E);
        normalize_n_kernel<<<ceil_div_u((long long)N * 64, TB), TB, 0, stream>>>(n_sum, o_sum, n_cnt,
                                                                                 (long long)N * 64);

        // ne_inner = leaky_relu(ne@Wi1 + n_agg@Wi2)
        gemm3_kernel<<<gemm_grid(N), TB, 0, stream>>>(ne, Wi1i, n_sum, Wi2i, nullptr, nullptr, ne_inner, N, 1);
        // ne_next = tanh(ne_inner@Wo_n + o_agg@Wo_e + n_agg@Wo_s)
        gemm3_kernel<<<gemm_grid(N), TB, 0, stream>>>(ne_inner, Won, o_sum, Woe, n_sum, Wos, ne_next, N, 2);

        float* t = ne; ne = ne_next; ne_next = t;
    }

    // ---- outputs: node_out (N x 64) then he (H x 64), concatenated ----
    float* out = (float*)d_out;
    hipMemcpyAsync(out, node_emb, (size_t)N * 64 * 4, hipMemcpyDeviceToDevice, stream);
    scatter_out_kernel<<<ceil_div_u((long long)NB * 64, TB), TB, 0, stream>>>(out, ne, node_indices, NB);
    hipMemcpyAsync(out + (size_t)N * 64, he, (size_t)H * 64 * 4, hipMemcpyDeviceToDevice, stream);
}